// Scene_10977936408973
// MI455X (gfx1250) — compile-verified
//
#include <hip/hip_runtime.h>
#include <math.h>

typedef __attribute__((ext_vector_type(2))) float v2f;
typedef __attribute__((ext_vector_type(8))) float v8f;

#define EPSH 1e-4f

// DPP16 lane swizzles (pure VALU, no LDS). Patterns chosen so the 4-step
// butterfly fully mixes each 16-lane half: xor1, xor2, half-mirror, mirror.
#define DPP_XOR1  0xB1   // quad_perm [1,0,3,2]
#define DPP_XOR2  0x4E   // quad_perm [2,3,0,1]
#define DPP_HMIR  0x141  // row_half_mirror (reverse within 8)
#define DPP_MIR   0x140  // row_mirror      (reverse within 16)

template <int CTRL>
__device__ __forceinline__ float dpp_swz_f(float x) {
    int i = __builtin_bit_cast(int, x);
    int r = __builtin_amdgcn_update_dpp(i, i, CTRL, 0xf, 0xf, true);
    return __builtin_bit_cast(float, r);
}
template <int CTRL>
__device__ __forceinline__ int dpp_swz_i(int x) {
    return __builtin_amdgcn_update_dpp(x, x, CTRL, 0xf, 0xf, true);
}

// One wave32 = 16 rays. Block of 256 threads = 8 waves = 128 rays.
// S = 128 surfaces = 8 WMMA tiles of 16.
__global__ __launch_bounds__(256) void raytrace_wmma(
    const float* __restrict__ pos, const float* __restrict__ dirs,
    const float* __restrict__ inten, const float* __restrict__ centers,
    const float* __restrict__ radii, const float* __restrict__ refl,
    float* __restrict__ out, int N)
{
    const int lane = threadIdx.x & 31;
    const int wave = threadIdx.x >> 5;
    const int lo   = lane & 15;
    const int hi   = lane >> 4;           // 0: K={0,1} half, 1: K={2,3} half
    const int rayBase = blockIdx.x * 128 + wave * 16;

    // ---- A matrices (16x4 f32). K slots: {dx,dy,dz,pd} and {px,py,pz,pq}
    // so WMMA directly yields b = pd - dirs.c and q = pq - 2*pos.c.
    int rA  = rayBase + lo;
    int rAc = rA < N ? rA : (N - 1);      // clamp tail loads; stores masked later
    const float* pp3 = pos  + 3 * rAc;
    const float* dd3 = dirs + 3 * rAc;
    float px = pp3[0], py = pp3[1], pz = pp3[2];
    float dx = dd3[0], dy = dd3[1], dz = dd3[2];
    float pd = px * dx + py * dy + pz * dz;   // pos.dir
    float pq = px * px + py * py + pz * pz;   // |pos|^2

    v2f a_dir, a_pos;
    a_dir[0] = hi ? dz : dx;
    a_dir[1] = hi ? pd : dy;
    a_pos[0] = hi ? pz : px;
    a_pos[1] = hi ? pq : py;

    // ---- B matrices (4x16 f32) for all 8 tiles.
    // B1 column s = (-cx,-cy,-cz, 1)   -> D1 = pd - dirs.c = b
    // B2 column s = (-2cx,-2cy,-2cz,1) -> D2 = pq - 2*pos.c
    // csv[t] = |c|^2 - r^2 for surface 16*t + lo (per-lane, no shuffle needed).
    v2f  Bm1[8], Bm2[8];
    float csv[8];
#pragma unroll
    for (int t = 0; t < 8; ++t) {
        int s = 16 * t + lo;
        const float* cp = centers + 3 * s;
        float cx = cp[0], cy = cp[1], cz = cp[2];
        float r  = radii[s];
        Bm1[t][0] = hi ? -cz : -cx;
        Bm1[t][1] = hi ? 1.0f : -cy;
        Bm2[t][0] = hi ? -2.0f * cz : -2.0f * cx;
        Bm2[t][1] = hi ? 1.0f : -2.0f * cy;
        csv[t] = cx * cx + cy * cy + cz * cz - r * r;
    }

    // ---- Main loop: 8 tiles x (2 WMMA + elementwise quadratic solve) ----
    float bestT[8];
    int   bestI[8];
#pragma unroll
    for (int v = 0; v < 8; ++v) { bestT[v] = __builtin_inff(); bestI[v] = 0; }

    const v8f czero = {};
#pragma unroll
    for (int t = 0; t < 8; ++t) {
        v8f Bq = __builtin_amdgcn_wmma_f32_16x16x4_f32(
            false, a_dir, false, Bm1[t], (short)0, czero, false, false);
        v8f Qq = __builtin_amdgcn_wmma_f32_16x16x4_f32(
            false, a_pos, false, Bm2[t], (short)0, czero, false, false);
        int sidx = 16 * t + lo;
#pragma unroll
        for (int v = 0; v < 8; ++v) {
            float b    = Bq[v];                      // (pos-c).dir
            float cc   = Qq[v] + csv[t];             // |pos-c|^2 - r^2
            float disc = fmaf(b, b, -cc);
            // Raw v_sqrt_f32: 1-instruction, quiet-NaN on disc<0 (marks invalid)
            float sq   = __builtin_amdgcn_sqrtf(disc);
            float t1   = -b - sq;
            float t2   = sq - b;
            float tt   = (t1 > EPSH) ? t1 : t2;      // NaN-safe: picks t2(=NaN)
            tt = (tt > EPSH) ? tt : __builtin_inff();// NaN/miss -> +inf
            if (tt < bestT[v]) { bestT[v] = tt; bestI[v] = sidx; }
        }
    }

    // ---- Cross-lane min + argmin over the 16 surface slots, DPP-only ----
    float redT[8];
    int   redI[8];
#pragma unroll
    for (int v = 0; v < 8; ++v) {
        float tm = bestT[v];                         // no NaNs here (inf or finite)
        tm = fminf(tm, dpp_swz_f<DPP_XOR1>(tm));
        tm = fminf(tm, dpp_swz_f<DPP_XOR2>(tm));
        tm = fminf(tm, dpp_swz_f<DPP_HMIR>(tm));
        tm = fminf(tm, dpp_swz_f<DPP_MIR >(tm));
        int cand = (bestT[v] == tm) ? bestI[v] : 0x7FFFFFFF;
        cand = min(cand, dpp_swz_i<DPP_XOR1>(cand));
        cand = min(cand, dpp_swz_i<DPP_XOR2>(cand));
        cand = min(cand, dpp_swz_i<DPP_HMIR>(cand));
        cand = min(cand, dpp_swz_i<DPP_MIR >(cand));
        redT[v] = tm;                                // first-min index, as jnp.argmin
        redI[v] = cand;
    }

    // ---- Epilogue: lane (lo==v, hi) handles ray v + 8*hi ----
    float myT = redT[0];
    int   myI = redI[0];
#pragma unroll
    for (int v = 1; v < 8; ++v) {
        if (lo == v) { myT = redT[v]; myI = redI[v]; }
    }
    int myRay = rayBase + (lane & 7) + 8 * hi;
    if (lo < 8 && myRay < N) {
        const float* p3 = pos  + 3 * myRay;
        const float* d3 = dirs + 3 * myRay;
        float p0 = p3[0], p1 = p3[1], p2 = p3[2];
        float d0 = d3[0], d1 = d3[1], d2 = d3[2];
        float ii = inten[myRay];

        bool hit = (myT < __builtin_inff()) && (ii > 0.0f);
        float mt = hit ? myT : 0.0f;

        const float* cw = centers + 3 * myI;
        float c0 = cw[0], c1 = cw[1], c2 = cw[2];
        float rw = radii[myI];
        float rf = refl[myI];

        float h0 = fmaf(mt, d0, p0);
        float h1 = fmaf(mt, d1, p1);
        float h2 = fmaf(mt, d2, p2);
        float inv_r = __builtin_amdgcn_rcpf(rw);     // single v_rcp_f32
        float n0 = (h0 - c0) * inv_r;
        float n1 = (h1 - c1) * inv_r;
        float n2 = (h2 - c2) * inv_r;
        float dn = d0 * n0 + d1 * n1 + d2 * n2;
        float s  = -2.0f * dn;
        float r0 = fmaf(s, n0, d0);
        float r1 = fmaf(s, n1, d1);
        float r2 = fmaf(s, n2, d2);
        float ni = ii * rf;

        float* o = out + 7 * myRay;
        o[0] = hit ? h0 : p0;
        o[1] = hit ? h1 : p1;
        o[2] = hit ? h2 : p2;
        o[3] = hit ? r0 : d0;
        o[4] = hit ? r1 : d1;
        o[5] = hit ? r2 : d2;
        o[6] = hit ? ni : ii;
    }
}

extern "C" void kernel_launch(void* const* d_in, const int* in_sizes, int n_in,
                              void* d_out, int out_size, void* d_ws, size_t ws_size,
                              hipStream_t stream) {
    const float* pos     = (const float*)d_in[0];
    const float* dirs    = (const float*)d_in[1];
    const float* inten   = (const float*)d_in[2];
    const float* centers = (const float*)d_in[3];
    const float* radii   = (const float*)d_in[4];
    const float* refl    = (const float*)d_in[5];
    float* out = (float*)d_out;

    int N = in_sizes[2];                 // intensity has one element per ray
    int blocks = (N + 127) / 128;        // 128 rays per 256-thread block
    raytrace_wmma<<<blocks, 256, 0, stream>>>(pos, dirs, inten, centers, radii,
                                              refl, out, N);
}